// MissingValueHandlerLayer_72773925864084
// MI455X (gfx1250) — compile-verified
//
#include <hip/hip_runtime.h>
#include <cstdint>

// x: (B, T, F) f32; out: (B, T, 2F) f32. missing <=> x == 0.0f.
#define B_   128
#define T_   2048
#define F_   64
#define FV   (F_ / 4)       // 16 float4 groups per row
#define CH   64             // timesteps per chunk
#define NCH  (T_ / CH)      // 32 chunks along T
#define SUB  16             // timesteps per sub-chunk inside k_fill
#define NTY  (CH / SUB)     // 4 sub-chunks

// Scan operator per component: run' = (v != 0) ? v : run. run==0 <=> no valid yet.
__device__ __forceinline__ void upd4(float4& run, const float4 v) {
  run.x = (v.x != 0.0f) ? v.x : run.x;
  run.y = (v.y != 0.0f) ? v.y : run.y;
  run.z = (v.z != 0.0f) ? v.z : run.z;
  run.w = (v.w != 0.0f) ? v.w : run.w;
}

// ---------------------------------------------------------------------------
// Kernel 1: chunk-local carries. Block 64 = (fx:16 float4-groups, c4:4 chunks),
// grid (NCH/4, B). Each thread streams its 64-timestep chunk with B128 loads
// (512B per wave per instruction) and records the last nonzero per component.
// ---------------------------------------------------------------------------
__global__ __launch_bounds__(64) void k_chunk_carry(const float* __restrict__ x,
                                                    float* __restrict__ carry) {
  const int tid = threadIdx.x;
  const int fx  = tid & (FV - 1);         // float4 group
  const int c   = blockIdx.x * 4 + (tid >> 4);
  const int b   = blockIdx.y;

  const float4* base =
      (const float4*)(x + ((size_t)b * T_ + (size_t)c * CH) * F_) + fx;

  float4 run = make_float4(0.f, 0.f, 0.f, 0.f);
  for (int tb = 0; tb < CH; tb += 16) {
    int tp = tb + 32; if (tp >= CH) tp = CH - 1;
    __builtin_prefetch(base + (size_t)tp * FV, 0, 0);   // global_prefetch_b8
#pragma unroll
    for (int j = 0; j < 16; ++j) {                      // 16-deep load clauses
      float4 v = base[(size_t)(tb + j) * FV];
      upd4(run, v);
    }
  }
  ((float4*)carry)[((size_t)b * NCH + c) * FV + fx] = run;
}

// ---------------------------------------------------------------------------
// Kernel 2: in-place exclusive scan of the NCH chunk carries per (b, f4).
// Block 64 = (fx:16, bsub:4), grid B/4. Tiny (1 MB of traffic).
// ---------------------------------------------------------------------------
__global__ __launch_bounds__(64) void k_scan_carry(float* __restrict__ carry) {
  const int tid = threadIdx.x;
  const int fx  = tid & (FV - 1);
  const int b   = blockIdx.x * 4 + (tid >> 4);
  float4* cw = (float4*)carry;

  float4 run = make_float4(0.f, 0.f, 0.f, 0.f);
#pragma unroll
  for (int c = 0; c < NCH; ++c) {
    size_t idx = ((size_t)b * NCH + c) * FV + fx;
    float4 v = cw[idx];
    cw[idx] = run;
    upd4(run, v);
  }
}

// ---------------------------------------------------------------------------
// Kernel 3: async-stage the (CH x F) tile into LDS (ASYNCcnt path), 2-level
// scan inside the tile, emit imputed + indicator with B128 stores.
// Block 64 = (fx:16, ty:4 sub-chunks of 16 t), grid (NCH, B) = 4096 blocks.
// ---------------------------------------------------------------------------
__global__ __launch_bounds__(64) void k_fill(const float* __restrict__ x,
                                             const float* __restrict__ carry,
                                             float* __restrict__ out) {
  __shared__ float4 tile[CH * FV];   // 16 KB
  __shared__ float4 subc[NTY * FV];  // 1 KB

  const int tid = threadIdx.x;
  const int fx  = tid & (FV - 1);
  const int ty  = tid >> 4;
  const int c   = blockIdx.x;
  const int b   = blockIdx.y;

  const float* gbase = x + ((size_t)b * T_ + (size_t)c * CH) * F_;

  // ---- async copy: CH*F_*4 = 16384 bytes, 16B per lane per op, 16 ops/thread.
  {
    const uint32_t lds0 = (uint32_t)(uintptr_t)(&tile[0]);  // flat addr[31:0] = LDS byte offset
#pragma unroll
    for (int k = 0; k < (CH * F_ * 4) / (64 * 16); ++k) {   // 16 iterations
      uint32_t byteoff = (uint32_t)(k * 64 + tid) * 16u;
      uint32_t ldst    = lds0 + byteoff;
      asm volatile("global_load_async_to_lds_b128 %0, %1, %2 offset:0"
                   :
                   : "v"(ldst), "v"(byteoff), "s"(gbase)
                   : "memory");
    }
    asm volatile("s_wait_asynccnt 0x0" ::: "memory");
  }
  __syncthreads();

  // ---- phase 1: sub-chunk local carries (ds_load_b128 per timestep)
  const int t0 = ty * SUB;
  float4 run = make_float4(0.f, 0.f, 0.f, 0.f);
#pragma unroll
  for (int j = 0; j < SUB; ++j) {
    float4 v = tile[(t0 + j) * FV + fx];
    upd4(run, v);
  }
  subc[ty * FV + fx] = run;
  __syncthreads();

  // ---- carry-in: global chunk carry-in folded with earlier sub-chunks
  float4 cin = ((const float4*)carry)[((size_t)b * NCH + c) * FV + fx];
#pragma unroll
  for (int yy = 0; yy < NTY - 1; ++yy) {
    float4 s = subc[yy * FV + fx];
    bool on = (yy < ty);
    cin.x = (on && s.x != 0.0f) ? s.x : cin.x;
    cin.y = (on && s.y != 0.0f) ? s.y : cin.y;
    cin.z = (on && s.z != 0.0f) ? s.z : cin.z;
    cin.w = (on && s.w != 0.0f) ? s.w : cin.w;
  }

  // ---- phase 2: emit imputed (== run after update) and indicators.
  run = cin;
  float4* obase = (float4*)(out + ((size_t)b * T_ + (size_t)c * CH) * (2 * F_));
#pragma unroll
  for (int j = 0; j < SUB; ++j) {
    int t = t0 + j;
    float4 v = tile[t * FV + fx];
    float4 ind;
    ind.x = (v.x == 0.0f) ? 1.0f : 0.0f;
    ind.y = (v.y == 0.0f) ? 1.0f : 0.0f;
    ind.z = (v.z == 0.0f) ? 1.0f : 0.0f;
    ind.w = (v.w == 0.0f) ? 1.0f : 0.0f;
    upd4(run, v);   // valid -> run=v; missing -> keep carry (0 if none yet)
    obase[(size_t)t * (2 * FV) + fx]      = run;   // imputed
    obase[(size_t)t * (2 * FV) + FV + fx] = ind;   // indicator
  }
}

// ---------------------------------------------------------------------------
extern "C" void kernel_launch(void* const* d_in, const int* in_sizes, int n_in,
                              void* d_out, int out_size, void* d_ws, size_t ws_size,
                              hipStream_t stream) {
  (void)in_sizes; (void)n_in; (void)out_size; (void)ws_size;
  const float* x   = (const float*)d_in[0];
  float*       out = (float*)d_out;
  float*       cw  = (float*)d_ws;   // needs B_*NCH*F_*4 = 1 MB scratch

  hipLaunchKernelGGL(k_chunk_carry, dim3(NCH / 4, B_), dim3(64), 0, stream, x, cw);
  hipLaunchKernelGGL(k_scan_carry,  dim3(B_ / 4),      dim3(64), 0, stream, cw);
  hipLaunchKernelGGL(k_fill,        dim3(NCH, B_),     dim3(64), 0, stream, x, cw, out);
}